// IHCEnvelope_4148938408288
// MI455X (gfx1250) — compile-verified
//
#include <hip/hip_runtime.h>
#include <cstdint>

// ---------------------------------------------------------------------------
// IHC envelope: y = rectify(x) -> 5x serial 1st-order IIR along T.
// Blocked parallel scan over the linear recurrence:
//   state s = (x0prev, y1, y2, y3, y4, y5)   (6-vector, affine per chunk)
//   pass1: per-chunk zero-state response c
//   pass2: per-channel scan  s[j] = A*s[j-1] + c[j-1],  A = M^L
//   pass3: per-chunk recurrence from correct initial state, write y5 (NT stores)
// Tiles staged through LDS with CDNA5 async global->LDS loads (ASYNCcnt),
// double-buffered per wave. LDS accessed via direct shared-array indexing so
// codegen uses ds_* ops (not flat).
// ---------------------------------------------------------------------------

static constexpr unsigned CH       = 4u * 31u;     // 124 channels
static constexpr unsigned T_LEN    = 441000u;      // samples per channel
static constexpr unsigned L_CHUNK  = 1024u;        // chunk length (2^10)
static constexpr unsigned CPC      = 431u;         // ceil(441000/1024)
static constexpr unsigned TOTALCK  = CH * CPC;     // 53444 chunks
static constexpr unsigned LAST_LEN = T_LEN - (CPC - 1u) * L_CHUNK; // 680
static constexpr unsigned ROWP     = 34u;          // LDS row pitch: 8B-aligned pairs, conflict-free
static constexpr unsigned TILE_F   = 32u * ROWP;   // floats per tile (1088)

typedef float v2f __attribute__((ext_vector_type(2)));

__device__ __forceinline__ void async_load_b64_to_lds(unsigned ldsByteOff, const void* gptr) {
    asm volatile("global_load_async_to_lds_b64 %0, %1, off"
                 :: "v"(ldsByteOff), "v"((unsigned long long)(uintptr_t)gptr)
                 : "memory");
}

// One IIR cascade step. s0 = rectified x[t-1]; s1..s5 = stage outputs at t-1.
__device__ __forceinline__ void iir_step(float in, float b0, float b1, float na1,
                                         float& s0, float& s1, float& s2,
                                         float& s3, float& s4, float& s5) {
    float u  = fmaxf(in, 0.0f);
    float y1 = fmaf(b0, u,  fmaf(b1, s0, na1 * s1));
    float y2 = fmaf(b0, y1, fmaf(b1, s1, na1 * s2));
    float y3 = fmaf(b0, y2, fmaf(b1, s2, na1 * s3));
    float y4 = fmaf(b0, y3, fmaf(b1, s3, na1 * s4));
    float y5 = fmaf(b0, y4, fmaf(b1, s4, na1 * s5));
    s0 = u; s1 = y1; s2 = y2; s3 = y3; s4 = y4; s5 = y5;
}

// WRITE_OUT=false : pass1 (emit per-chunk zero-state response c)
// WRITE_OUT=true  : pass3 (start from sInit, write outputs)
template <bool WRITE_OUT>
__global__ __launch_bounds__(128) void iir_chunk_pass(const float* __restrict__ x,
                                                      float* __restrict__ y,
                                                      const float* __restrict__ bco,
                                                      const float* __restrict__ aco,
                                                      float* __restrict__ cArr,
                                                      const float* __restrict__ sInitArr) {
    const unsigned tid  = threadIdx.x;
    const unsigned lane = tid & 31u;
    const unsigned wave = tid >> 5;
    const unsigned wc0  = (blockIdx.x * 4u + wave) * 32u;   // first chunk of this wave
    const unsigned myChunk = wc0 + lane;
    const bool laneValid = myChunk < TOTALCK;

    const float a0  = aco[0];
    const float b0  = bco[0] / a0;
    const float b1  = bco[1] / a0;
    const float na1 = -(aco[1] / a0);

    __align__(16) __shared__ float tiles[4u * 2u * TILE_F];   // 4 waves x 2 bufs
    const unsigned tb0 = (wave * 2u + 0u) * TILE_F;           // element offsets into tiles[]
    const unsigned tb1 = (wave * 2u + 1u) * TILE_F;
    const unsigned ldsBase = (unsigned)(uintptr_t)&tiles[0];  // LDS byte address of array base

    // recurrence state
    float s0 = 0.f, s1 = 0.f, s2 = 0.f, s3 = 0.f, s4 = 0.f, s5 = 0.f;
    if (WRITE_OUT && laneValid) {
        const float* si = sInitArr + (size_t)myChunk * 6u;
        s0 = si[0]; s1 = si[1]; s2 = si[2]; s3 = si[3]; s4 = si[4]; s5 = si[5];
    }

    // fast-path eligibility (wave-uniform)
    const bool allValid   = (wc0 + 31u) < TOTALCK;
    const unsigned rmod   = wc0 % CPC;
    const bool hasPartial = rmod >= (CPC - 32u);             // includes chunk j==430
    const int nFast = allValid ? (hasPartial ? 21 : 32) : 0; // subtile 21+ touches tail

    // per-lane global element offsets for cooperative b64 tile transfers:
    // issue i covers rows 2i..2i+1; lane -> (row = 2i + lane/16, colpair = 2*(lane&15))
    unsigned gofs[16];
#pragma unroll
    for (int i = 0; i < 16; ++i) gofs[i] = 0u;
    if (allValid) {
#pragma unroll
        for (int i = 0; i < 16; ++i) {
            unsigned rc = wc0 + 2u * (unsigned)i + (lane >> 4);
            unsigned ch = rc / CPC;
            unsigned j  = rc - ch * CPC;
            gofs[i] = ch * T_LEN + j * L_CHUNK + ((lane & 15u) << 1);
        }
    }

    auto issueTile = [&](unsigned tb, unsigned t0) {
#pragma unroll
        for (int i = 0; i < 16; ++i) {
            unsigned elem = tb + (2u * (unsigned)i + (lane >> 4)) * ROWP + ((lane & 15u) << 1);
            async_load_b64_to_lds(ldsBase + (elem << 2), x + gofs[i] + t0);
        }
    };

    if (nFast > 0) issueTile(tb0, 0u);   // prologue

    for (int st = 0; st < 32; ++st) {
        const unsigned t0 = (unsigned)st * 32u;
        const unsigned tb  = (st & 1) ? tb1 : tb0;
        const unsigned tbn = (st & 1) ? tb0 : tb1;

        if (st < nFast) {
            // ---- fast path: async double-buffered tiles ----
            if (st + 1 < nFast) {
                issueTile(tbn, t0 + 32u);
                asm volatile("s_wait_asynccnt 16" ::: "memory"); // this tile's 16 ops done
            } else {
                asm volatile("s_wait_asynccnt 0" ::: "memory");
            }
            const unsigned rowb = tb + lane * ROWP;
#pragma unroll
            for (int t = 0; t < 32; ++t) {
                iir_step(tiles[rowb + (unsigned)t], b0, b1, na1, s0, s1, s2, s3, s4, s5);
                if (WRITE_OUT) tiles[rowb + (unsigned)t] = s5;
            }
            if (WRITE_OUT) {
                asm volatile("" ::: "memory");
#pragma unroll
                for (int i = 0; i < 16; ++i) {
                    unsigned e = tb + (2u * (unsigned)i + (lane >> 4)) * ROWP + ((lane & 15u) << 1);
                    v2f v = *reinterpret_cast<const v2f*>(&tiles[e]);
                    __builtin_nontemporal_store(v, reinterpret_cast<v2f*>(y + gofs[i] + t0));
                }
            }
        } else {
            // ---- slow path: edge waves / channel-tail subtiles ----
            for (int k = 0; k < 32; ++k) {
                unsigned rc = wc0 + (unsigned)k;
                float v = 0.0f;
                if (rc < TOTALCK) {
                    unsigned ch  = rc / CPC;
                    unsigned j   = rc - ch * CPC;
                    unsigned len = (j == CPC - 1u) ? LAST_LEN : L_CHUNK;
                    unsigned idx = t0 + lane;
                    if (idx < len) v = x[ch * T_LEN + j * L_CHUNK + idx];
                }
                tiles[tb + (unsigned)k * ROWP + lane] = v;
            }
            asm volatile("" ::: "memory");
            const unsigned rowb = tb + lane * ROWP;
#pragma unroll
            for (int t = 0; t < 32; ++t) {
                iir_step(tiles[rowb + (unsigned)t], b0, b1, na1, s0, s1, s2, s3, s4, s5);
                if (WRITE_OUT) tiles[rowb + (unsigned)t] = s5;
            }
            asm volatile("" ::: "memory");
            if (WRITE_OUT) {
                for (int k = 0; k < 32; ++k) {
                    unsigned rc = wc0 + (unsigned)k;
                    if (rc < TOTALCK) {
                        unsigned ch  = rc / CPC;
                        unsigned j   = rc - ch * CPC;
                        unsigned len = (j == CPC - 1u) ? LAST_LEN : L_CHUNK;
                        unsigned idx = t0 + lane;
                        if (idx < len)
                            __builtin_nontemporal_store(tiles[tb + (unsigned)k * ROWP + lane],
                                                        &y[ch * T_LEN + j * L_CHUNK + idx]);
                    }
                }
            }
        }
    }

    if (!WRITE_OUT && laneValid) {
        float* c = cArr + (size_t)myChunk * 6u;
        c[0] = s0; c[1] = s1; c[2] = s2; c[3] = s3; c[4] = s4; c[5] = s5;
    }
}

// Pass 2: per-channel affine scan over chunk carries. A = M^1024 (shared by all
// full chunks); computed per thread via 10 squarings of the 6x6 companion map.
__global__ __launch_bounds__(128) void iir_carry_scan(const float* __restrict__ bco,
                                                      const float* __restrict__ aco,
                                                      const float* __restrict__ cArr,
                                                      float* __restrict__ sInitArr) {
    const unsigned ch = blockIdx.x * blockDim.x + threadIdx.x;
    if (ch >= CH) return;

    const float a0  = aco[0];
    const float b0  = bco[0] / a0;
    const float b1  = bco[1] / a0;
    const float na1 = -(aco[1] / a0);

    // Single-step homogeneous propagator M (row 0 is zero: x0' comes from input)
    float M[6][6];
    for (int i = 0; i < 6; ++i)
        for (int j = 0; j < 6; ++j) M[i][j] = 0.0f;
    M[1][0] = b1; M[1][1] = na1;
    for (int k = 2; k < 6; ++k) {
        for (int j = 0; j < 6; ++j) M[k][j] = b0 * M[k - 1][j];
        M[k][k - 1] += b1;
        M[k][k]     += na1;
    }
    // A = M^1024
    float A[6][6], Tm[6][6];
    for (int i = 0; i < 6; ++i)
        for (int j = 0; j < 6; ++j) A[i][j] = M[i][j];
    for (int it = 0; it < 10; ++it) {
        for (int i = 0; i < 6; ++i)
            for (int j = 0; j < 6; ++j) {
                float acc = 0.0f;
                for (int k = 0; k < 6; ++k) acc = fmaf(A[i][k], A[k][j], acc);
                Tm[i][j] = acc;
            }
        for (int i = 0; i < 6; ++i)
            for (int j = 0; j < 6; ++j) A[i][j] = Tm[i][j];
    }

    const float* cRow = cArr + (size_t)ch * CPC * 6u;
    float* oRow       = sInitArr + (size_t)ch * CPC * 6u;

    float s[6] = {0.f, 0.f, 0.f, 0.f, 0.f, 0.f};
    for (int i = 0; i < 6; ++i) oRow[i] = 0.0f;       // chunk 0: zero initial state
    for (unsigned j = 1; j < CPC; ++j) {
        const float* cv = cRow + (size_t)(j - 1) * 6u;
        float ns[6];
        for (int i = 0; i < 6; ++i) {
            float acc = cv[i];
            for (int k = 0; k < 6; ++k) acc = fmaf(A[i][k], s[k], acc);
            ns[i] = acc;
        }
        float* o = oRow + (size_t)j * 6u;
        for (int i = 0; i < 6; ++i) { s[i] = ns[i]; o[i] = ns[i]; }
    }
}

extern "C" void kernel_launch(void* const* d_in, const int* in_sizes, int n_in,
                              void* d_out, int out_size, void* d_ws, size_t ws_size,
                              hipStream_t stream) {
    (void)in_sizes; (void)n_in; (void)out_size; (void)ws_size;
    const float* x = (const float*)d_in[0];
    const float* b = (const float*)d_in[1];
    const float* a = (const float*)d_in[2];
    float* y = (float*)d_out;

    float* cArr     = (float*)d_ws;                       // TOTALCK * 6 floats
    float* sInitArr = cArr + (size_t)TOTALCK * 6u;        // TOTALCK * 6 floats

    const unsigned nWaves  = (TOTALCK + 31u) / 32u;       // 1671
    const unsigned nBlocks = (nWaves + 3u) / 4u;          // 418 blocks of 128 (4 waves)

    iir_chunk_pass<false><<<nBlocks, 128, 0, stream>>>(x, y, b, a, cArr, sInitArr);
    iir_carry_scan<<<1, 128, 0, stream>>>(b, a, cArr, sInitArr);
    iir_chunk_pass<true><<<nBlocks, 128, 0, stream>>>(x, y, b, a, cArr, sInitArr);
}